// DeltaRuleMemory_35270271435153
// MI455X (gfx1250) — compile-verified
//
#include <hip/hip_runtime.h>
#include <hip/hip_bf16.h>

// ---------------------------------------------------------------------------
// DeltaRule fast-weight memory, B=32, L=2048, H=512, fp32.
//
// Kernel 1 (delta_rows): one wave32 per (batch, row-of-M). The 512-float row
//   state lives in 16 VGPRs/lane (4 x float4) for the entire 2047-step scan.
//   Per step: coalesced b128 loads of k, fused dot(m,k) + dot(k,k), wave32
//   xor-reduction, scalar k[row], rank-1 axpy. Final context = dot(m, query).
// Kernel 2 (proj_wmma): context @ W^T + bias via V_WMMA_F32_16X16X4_F32
//   (fp32 WMMA keeps full precision). 64 tiles, 1 wave each, K=512 in steps
//   of 4.
// ---------------------------------------------------------------------------

namespace {
constexpr int kB = 32;
constexpr int kL = 2048;
constexpr int kH = 512;
constexpr float kEps = 1e-6f;
}  // namespace

typedef __attribute__((ext_vector_type(2))) float v2f;
typedef __attribute__((ext_vector_type(8))) float v8f;

__device__ __forceinline__ float wave_sum(float x) {
#pragma unroll
  for (int off = 16; off > 0; off >>= 1) x += __shfl_xor(x, off, 32);
  return x;  // all 32 lanes hold the sum
}

#define DOT4(acc, a, b)                                            \
  acc = fmaf((a).x, (b).x,                                         \
        fmaf((a).y, (b).y, fmaf((a).z, (b).z, fmaf((a).w, (b).w, acc))))

#define AXPY4(m, c, k)                                             \
  (m).x = fmaf((c), (k).x, (m).x);                                 \
  (m).y = fmaf((c), (k).y, (m).y);                                 \
  (m).z = fmaf((c), (k).z, (m).z);                                 \
  (m).w = fmaf((c), (k).w, (m).w)

__global__ __launch_bounds__(256, 4) void delta_rows(
    const float* __restrict__ hidden, float* __restrict__ ctx) {
  const int lane = threadIdx.x & 31;
  const int wave = blockIdx.x * (blockDim.x >> 5) + (threadIdx.x >> 5);
  const int b   = wave >> 9;          // wave / 512
  const int row = wave & (kH - 1);    // which row of M this wave owns

  const float* kptr = hidden + (size_t)b * kL * kH;  // hidden[b, 0, :]

  // Row state m_row[0..511]; lane j holds floats {128*q + 4*j + e}.
  float4 m0 = {0.f, 0.f, 0.f, 0.f}, m1 = m0, m2 = m0, m3 = m0;

#pragma unroll 1
  for (int t = 0; t < kL - 1; ++t) {
    const float4* k4 = (const float4*)kptr;
    float4 k0 = k4[lane];
    float4 k1 = k4[lane + 32];
    float4 k2 = k4[lane + 64];
    float4 k3 = k4[lane + 96];
    // Prefetch next timestep's k row (one byte per 64B line across lanes).
    __builtin_prefetch(kptr + kH + 16 * lane, 0, 1);

    float dot = 0.f, nrm = 0.f;
    DOT4(dot, m0, k0); DOT4(dot, m1, k1); DOT4(dot, m2, k2); DOT4(dot, m3, k3);
    DOT4(nrm, k0, k0); DOT4(nrm, k1, k1); DOT4(nrm, k2, k2); DOT4(nrm, k3, k3);
    dot = wave_sum(dot);
    nrm = wave_sum(nrm);

    const float ki = kptr[row];                 // uniform -> s_load
    const float c  = ki - dot / (nrm + kEps);   // delta_v[row]

    AXPY4(m0, c, k0); AXPY4(m1, c, k1); AXPY4(m2, c, k2); AXPY4(m3, c, k3);
    kptr += kH;
  }

  // kptr == hidden[b, L-1, :] (query); context[b,row] = dot(m_row, query)
  const float4* q4 = (const float4*)kptr;
  float4 q0 = q4[lane], q1 = q4[lane + 32], q2 = q4[lane + 64], q3 = q4[lane + 96];
  float dot = 0.f;
  DOT4(dot, m0, q0); DOT4(dot, m1, q1); DOT4(dot, m2, q2); DOT4(dot, m3, q3);
  dot = wave_sum(dot);
  if (lane == 0) ctx[(size_t)b * kH + row] = dot;
}

// out[bb, j] = sum_h ctx[bb, h] * W[j, h] + bias[j]
// One wave per 16x16 output tile; fp32 WMMA, K stepped by 4.
__global__ __launch_bounds__(32) void proj_wmma(
    const float* __restrict__ ctx, const float* __restrict__ W,
    const float* __restrict__ bias, float* __restrict__ out) {
  const int lane  = threadIdx.x & 31;
  const int tileN = blockIdx.x & 31;   // output-column tile (j)
  const int tileM = blockIdx.x >> 5;   // batch-row tile (bb), 0..1
  const int r16   = lane & 15;
  const int hi    = lane >> 4;         // 0: K={0,1}; 1: K={2,3}

  // A (16x4 f32): lane r16<16 rows, K pair selected by `hi`.
  const float* arow = ctx + (size_t)(tileM * 16 + r16) * kH + 2 * hi;
  // B (4x16 f32): B[k][n] = W[tileN*16+n][kk+k]; lane holds column n=r16.
  const float* brow = W + (size_t)(tileN * 16 + r16) * kH + 2 * hi;

  v8f acc = {};
#pragma unroll 8
  for (int kk = 0; kk < kH; kk += 4) {
    v2f a = *(const v2f*)(arow + kk);   // global_load_b64
    v2f bm = *(const v2f*)(brow + kk);  // global_load_b64
    acc = __builtin_amdgcn_wmma_f32_16x16x4_f32(
        /*neg_a=*/false, a, /*neg_b=*/false, bm,
        /*c_mod=*/(short)0, acc, /*reuse_a=*/false, /*reuse_b=*/false);
  }

  // D layout: vgpr r, lane L -> row (r + 8*(L>=16)), col (L&15).
  const int j  = tileN * 16 + r16;
  const float bj = bias[j];
  const int row0 = tileM * 16 + 8 * hi;
#pragma unroll
  for (int r = 0; r < 8; ++r)
    out[(size_t)(row0 + r) * kH + j] = acc[r] + bj;
}

extern "C" void kernel_launch(void* const* d_in, const int* in_sizes, int n_in,
                              void* d_out, int out_size, void* d_ws, size_t ws_size,
                              hipStream_t stream) {
  const float* hidden = (const float*)d_in[0];  // (32, 2048, 512) fp32
  const float* W      = (const float*)d_in[1];  // (512, 512) fp32
  const float* bias   = (const float*)d_in[2];  // (512,) fp32
  float* out = (float*)d_out;                   // (32, 512) fp32
  float* ctx = (float*)d_ws;                    // (32, 512) fp32 scratch

  // 32*512 = 16384 waves, 8 waves (256 threads) per block -> 2048 blocks.
  delta_rows<<<dim3((kB * kH) / 8), dim3(256), 0, stream>>>(hidden, ctx);
  // 2 batch-tiles x 32 column-tiles = 64 single-wave blocks.
  proj_wmma<<<dim3(64), dim3(32), 0, stream>>>(ctx, W, bias, out);
}